// EnhancedMPOTransformer_60404420051680
// MI455X (gfx1250) — compile-verified
//
#include <hip/hip_runtime.h>

typedef float v2f __attribute__((ext_vector_type(2)));
typedef float v4f __attribute__((ext_vector_type(4)));
typedef float v8f __attribute__((ext_vector_type(8)));

#define TSP      200
#define NPAIR    800          // 4 traj * 200 steps: all distinct results
#define OUT_PER  3168         // 528 * 6 floats per (traj,tin)
#define N2       297          // 33 * 9 columns of the middle matmul
#define N2PAD    304          // padded LDS stride

// ---------------------------------------------------------------------------
// Kernel 1: one block per (traj, tin). Computes the full contraction chain
// for that pair and stores 3168 floats into ws[pair].
//   Stage 1: M1[16x24] = T1(4x8) @ T2(8x96)           (VALU, tiny)
//   Stage 2: P [16x297] = M1(16x24) @ T3(24x297)      (v_wmma_f32_16x16x4_f32)
//   Stage 3: out[528x6] = P(as 528x9) @ T4(9x6)       (VALU, tiny)
// ---------------------------------------------------------------------------
__global__ __launch_bounds__(256)
void mpo_compute_all(const float* __restrict__ T1_all,   // [4,4,1,8]
                     const float* __restrict__ T2_all,   // [4,8,4,1,24]
                     const float* __restrict__ T3_all,   // [4,200,24,33,1,9]
                     const float* __restrict__ T4_all,   // [4,9,3,2]
                     float* __restrict__ ws_out) {       // [800, 3168]
  __shared__ float sM1[16 * 24];
  __shared__ float sT4[64];
  __shared__ float sP[16 * N2PAD];

  const int pair = blockIdx.x;
  const int traj = pair / TSP;
  const int tid  = threadIdx.x;

  // ---- Stage 1: M1[i*4+p][w] = sum_d T1[traj,i,0,d] * T2[traj,d,p,0,w]
  const float* T1 = T1_all + traj * 32;    // [4][8]
  const float* T2 = T2_all + traj * 768;   // [8][4][24]
  for (int t = tid; t < 384; t += 256) {
    const int row = t / 24, w = t % 24;
    const int i = row >> 2, p = row & 3;
    float s = 0.f;
#pragma unroll
    for (int d = 0; d < 8; ++d)
      s += T1[i * 8 + d] * T2[d * 96 + p * 24 + w];
    sM1[t] = s;
  }
  if (tid < 54) sT4[tid] = T4_all[traj * 54 + tid];   // [9][6]
  __syncthreads();

  // ---- Stage 2: P = M1 @ T3 via f32 WMMA 16x16x4, fp32 throughout.
  const float* T3 = T3_all + (size_t)pair * (24 * N2);  // [24][297] row-major
  const int lane  = tid & 31;
  const int wv    = tid >> 5;        // 8 waves per block
  const int laneM = lane & 15;       // M (for A) / N (for B) index
  const int khalf = lane >> 4;       // 0 -> K{0,1}; 1 -> K{2,3}

  // A fragments for all 6 K-chunks (identical across this block's waves).
  // ISA layout (16x4 f32 A): VGPR0 = K(2*khalf), VGPR1 = K(2*khalf+1).
  v2f afrag[6];
#pragma unroll
  for (int kc = 0; kc < 6; ++kc) {
    afrag[kc].x = sM1[laneM * 24 + kc * 4 + 2 * khalf + 0];
    afrag[kc].y = sM1[laneM * 24 + kc * 4 + 2 * khalf + 1];
  }

  // 19 column tiles of 16, round-robin on waves. Tail columns 297..303:
  // clamp the load index instead of predicating — B column n only feeds
  // output column n, and those columns are never stored, so duplicated
  // (in-bounds) data is harmless. This keeps the load loop free of
  // exec-mask save/restore churn.
  for (int n0 = wv * 16; n0 < 304; n0 += 128) {
    const int  col  = n0 + laneM;
    const bool ok   = col < N2;
    const int  colc = ok ? col : (N2 - 1);   // clamped, always in-bounds
    v8f c = {};
#pragma unroll
    for (int kc = 0; kc < 6; ++kc) {
      const int k = kc * 4 + 2 * khalf;
      v2f b;
      b.x = T3[(k + 0) * N2 + colc];
      b.y = T3[(k + 1) * N2 + colc];
      c = __builtin_amdgcn_wmma_f32_16x16x4_f32(false, afrag[kc], false, b,
                                                (short)0, c, false, false);
    }
    if (ok) {
      // C/D layout: VGPR r holds row r (lanes 0-15) / row r+8 (lanes 16-31).
#pragma unroll
      for (int r = 0; r < 8; ++r)
        sP[(r + 8 * khalf) * N2PAD + col] = c[r];
    }
  }
  __syncthreads();

  // ---- Stage 3: out[r*6+m] = sum_w P[i][p*9+w] * T4[w][m],  r = i*33+p
  float* dst = ws_out + (size_t)pair * OUT_PER;
  for (int t = tid; t < OUT_PER; t += 256) {
    const int r = t / 6, m = t % 6;
    const int i = r / 33, p = r % 33;
    const float* prow = &sP[i * N2PAD + p * 9];
    float s = 0.f;
#pragma unroll
    for (int w = 0; w < 9; ++w)
      s += prow[w] * sT4[w * 6 + m];
    dst[t] = s;
  }
}

// ---------------------------------------------------------------------------
// Kernel 2: pure bandwidth. One block per sample: broadcast the precomputed
// 3168-float row (L2-resident, ~10 MB total) into the 207 MB output with
// 128-bit non-temporal stores.
// ---------------------------------------------------------------------------
__global__ __launch_bounds__(256)
void mpo_gather(const int* __restrict__ time_indices,
                const float* __restrict__ ws_out,
                float* __restrict__ out) {
  const int b   = blockIdx.x;
  const int idx = time_indices[b];                       // uniform -> scalar load
  const v4f* __restrict__ src = (const v4f*)(ws_out + (size_t)idx * OUT_PER);
  v4f* __restrict__ dst       = (v4f*)(out + (size_t)b * OUT_PER);
#pragma unroll 4
  for (int i = threadIdx.x; i < OUT_PER / 4; i += 256) { // 792 v4f's
    v4f v = src[i];
    __builtin_nontemporal_store(v, &dst[i]);
  }
}

extern "C" void kernel_launch(void* const* d_in, const int* in_sizes, int n_in,
                              void* d_out, int out_size, void* d_ws, size_t ws_size,
                              hipStream_t stream) {
  const int*   tidx = (const int*)  d_in[0];  // time_indices [16384] int32
  const float* T1   = (const float*)d_in[1];  // [4,4,1,8]
  const float* T2   = (const float*)d_in[2];  // [4,8,4,1,24]
  const float* T3   = (const float*)d_in[3];  // [4,200,24,33,1,9]
  const float* T4   = (const float*)d_in[4];  // [4,9,3,2]
  float* ws  = (float*)d_ws;                  // needs 800*3168*4 B ~= 9.7 MB
  float* out = (float*)d_out;                 // [16384, 3168]
  const int batch = in_sizes[0];

  mpo_compute_all<<<NPAIR, 256, 0, stream>>>(T1, T2, T3, T4, ws);
  mpo_gather<<<batch, 256, 0, stream>>>(tidx, ws, out);
}